// NGCFConv_62801011802126
// MI455X (gfx1250) — compile-verified
//
#include <hip/hip_runtime.h>

#define N_NODES 50000
#define N_EDGES 800000
#define F 64

typedef __attribute__((ext_vector_type(2))) float v2f;
typedef __attribute__((ext_vector_type(8))) float v8f;

// ---------------------------------------------------------------------------
// 1) Degree accumulation (atomics into zeroed buffers)
// ---------------------------------------------------------------------------
__global__ __launch_bounds__(256) void deg_kernel(const int* __restrict__ src,
                                                  const int* __restrict__ dst,
                                                  float* __restrict__ deg_out,
                                                  float* __restrict__ deg_in) {
    int e = blockIdx.x * 256 + threadIdx.x;
    if (e < N_EDGES) {
        atomicAdd(&deg_out[src[e]], 1.0f);
        atomicAdd(&deg_in[dst[e]], 1.0f);
    }
}

// ---------------------------------------------------------------------------
// 2) In-place deg -> 1/sqrt(max(deg,1)) for both deg arrays (contiguous 2N)
// ---------------------------------------------------------------------------
__global__ __launch_bounds__(256) void rsqrt_kernel(float* __restrict__ d) {
    int i = blockIdx.x * 256 + threadIdx.x;
    if (i < 2 * N_NODES) d[i] = rsqrtf(fmaxf(d[i], 1.0f));
}

// ---------------------------------------------------------------------------
// 3) feat_n = feature * norm_out   (one thread per 4 features)
// ---------------------------------------------------------------------------
__global__ __launch_bounds__(256) void featn_kernel(const float* __restrict__ feature,
                                                    const float* __restrict__ norm_out,
                                                    float* __restrict__ featn) {
    int t = blockIdx.x * 256 + threadIdx.x;   // N*16 threads
    int i = t >> 4;
    int c = (t & 15) << 2;
    if (i < N_NODES) {
        float s = norm_out[i];
        const float4 f = *(const float4*)(feature + i * F + c);
        float4 o;
        o.x = f.x * s; o.y = f.y * s; o.z = f.z * s; o.w = f.w * s;
        *(float4*)(featn + i * F + c) = o;
    }
}

// ---------------------------------------------------------------------------
// 4) Edge scatter: h_self[v] += fn[u]; h_inter[v] += fn[u]*fn[v]
//    16 lanes per edge, float4 per lane, global_atomic_add_f32
// ---------------------------------------------------------------------------
__global__ __launch_bounds__(256) void scatter_kernel(const int* __restrict__ src,
                                                      const int* __restrict__ dst,
                                                      const float* __restrict__ featn,
                                                      float* __restrict__ h_self,
                                                      float* __restrict__ h_inter) {
    int t = blockIdx.x * 256 + threadIdx.x;   // E*16 threads
    int e = t >> 4;
    int c = (t & 15) << 2;
    if (e >= N_EDGES) return;
    int u = src[e], v = dst[e];
    const float4 fu = *(const float4*)(featn + u * F + c);
    const float4 fv = *(const float4*)(featn + v * F + c);
    float* hs = h_self + v * F + c;
    float* hi = h_inter + v * F + c;
    atomicAdd(hs + 0, fu.x); atomicAdd(hs + 1, fu.y);
    atomicAdd(hs + 2, fu.z); atomicAdd(hs + 3, fu.w);
    atomicAdd(hi + 0, fu.x * fv.x); atomicAdd(hi + 1, fu.y * fv.y);
    atomicAdd(hi + 2, fu.z * fv.z); atomicAdd(hi + 3, fu.w * fv.w);
}

// ---------------------------------------------------------------------------
// 5) Dual GEMM + epilogue via V_WMMA_F32_16X16X4_F32 (fp32 tensor op).
//    One wave per 16-row tile (N % 16 == 0, EXEC all-1s as WMMA requires).
//    out = ((h_self@Ws + h_inter@Wi) * norm_in + feature) * 0.5
// ---------------------------------------------------------------------------
__global__ __launch_bounds__(256) void gemm_kernel(const float* __restrict__ h_self,
                                                   const float* __restrict__ h_inter,
                                                   const float* __restrict__ W_self,
                                                   const float* __restrict__ W_inter,
                                                   const float* __restrict__ norm_in,
                                                   const float* __restrict__ feature,
                                                   float* __restrict__ out) {
    const int lane = threadIdx.x & 31;
    const int wave = threadIdx.x >> 5;
    const int gw   = blockIdx.x * 8 + wave;
    const int row0 = gw * 16;
    if (row0 >= N_NODES) return;              // wave-uniform branch; EXEC stays full

    const int half = lane >> 4;               // 0: K pair {k,k+1}; 1: {k+2,k+3}
    const int l16  = lane & 15;
    const int arow = row0 + l16;              // A-matrix row for this lane

    v8f acc0 = {}, acc1 = {}, acc2 = {}, acc3 = {};

#pragma unroll
    for (int k = 0; k < F; k += 4) {
        const int kk = k + 2 * half;
        // A fragments: contiguous float2 within the row (8B aligned, kk even)
        v2f a_s = { h_self [arow * F + kk], h_self [arow * F + kk + 1] };
        v2f a_i = { h_inter[arow * F + kk], h_inter[arow * F + kk + 1] };
#pragma unroll
        for (int n = 0; n < 4; ++n) {
            const int col = n * 16 + l16;
            // B fragments: same column, two consecutive K rows (stride F)
            v2f b_s = { W_self [kk * F + col], W_self [(kk + 1) * F + col] };
            v2f b_i = { W_inter[kk * F + col], W_inter[(kk + 1) * F + col] };
            v8f& acc = (n == 0 ? acc0 : n == 1 ? acc1 : n == 2 ? acc2 : acc3);
            acc = __builtin_amdgcn_wmma_f32_16x16x4_f32(
                false, a_s, false, b_s, (short)0, acc, false, false);
            acc = __builtin_amdgcn_wmma_f32_16x16x4_f32(
                false, a_i, false, b_i, (short)0, acc, false, false);
        }
    }

    // Epilogue: C/D layout — VGPR r: lanes 0-15 -> M=r, lanes 16-31 -> M=r+8
#pragma unroll
    for (int n = 0; n < 4; ++n) {
        const v8f acc = (n == 0 ? acc0 : n == 1 ? acc1 : n == 2 ? acc2 : acc3);
        const int col = n * 16 + l16;
#pragma unroll
        for (int r = 0; r < 8; ++r) {
            const int m = row0 + r + 8 * half;
            out[m * F + col] = (acc[r] * norm_in[m] + feature[m * F + col]) * 0.5f;
        }
    }
}

// ---------------------------------------------------------------------------
extern "C" void kernel_launch(void* const* d_in, const int* in_sizes, int n_in,
                              void* d_out, int out_size, void* d_ws, size_t ws_size,
                              hipStream_t stream) {
    const float* feature = (const float*)d_in[0];
    const int*   src     = (const int*)d_in[1];
    const int*   dst     = (const int*)d_in[2];
    const float* W_self  = (const float*)d_in[3];
    const float* W_inter = (const float*)d_in[4];
    float*       out     = (float*)d_out;

    // Workspace layout (floats): [deg/norm_out N][deg/norm_in N][featn N*F][h_self N*F][h_inter N*F]
    float* ws      = (float*)d_ws;
    float* deg_out = ws;                         // becomes norm_out in-place
    float* deg_in  = ws + N_NODES;               // becomes norm_in in-place
    float* featn   = ws + 2 * N_NODES;
    float* h_self  = featn + (size_t)N_NODES * F;
    float* h_inter = h_self + (size_t)N_NODES * F;

    // Zero the atomic targets (capture-safe memset nodes)
    hipMemsetAsync(deg_out, 0, 2ull * N_NODES * sizeof(float), stream);
    hipMemsetAsync(h_self, 0, 2ull * (size_t)N_NODES * F * sizeof(float), stream);

    deg_kernel<<<(N_EDGES + 255) / 256, 256, 0, stream>>>(src, dst, deg_out, deg_in);
    rsqrt_kernel<<<(2 * N_NODES + 255) / 256, 256, 0, stream>>>(deg_out);
    featn_kernel<<<(N_NODES * 16 + 255) / 256, 256, 0, stream>>>(feature, deg_out, featn);
    scatter_kernel<<<(N_EDGES * 16 + 255) / 256, 256, 0, stream>>>(src, dst, featn, h_self, h_inter);

    // 3125 row tiles, 8 waves (256 threads) per block
    gemm_kernel<<<(N_NODES / 16 + 7) / 8, 256, 0, stream>>>(h_self, h_inter, W_self, W_inter,
                                                            deg_in, feature, out);
}